// Loss_90142773608781
// MI455X (gfx1250) — compile-verified
//
#include <hip/hip_runtime.h>
#include <hip/hip_bf16.h>

#define LAMDA_COORD 5.0f
#define LAMDA_NOOBJ 0.5f
#define LAMDA_OBJ   1.0f
#define LAMDA_SIZE  5.0f

#define GRID_N   7
#define BATCH_N  16384
#define NTGT_N   65536
#define CELLS    (BATCH_N * GRID_N * GRID_N)   // 802816
#define NFLOAT   (CELLS * 30)                  // 24084480 = 2^15 * 3 * 5 * 7^2
#define NV4      (NFLOAT / 4)                  // 6021120

#define THREADS    256
#define NBLK_CONF  1176                        // 1176*256 = 301056 threads
#define CONF_ITERS (NV4 / (NBLK_CONF * THREADS))  // exactly 20, no tail
#define CONF_OUTER (CONF_ITERS / 4)            // 5 outer iters, 4 loads each
#define NBLK_TGT   (NTGT_N / THREADS)          // 256

typedef __attribute__((ext_vector_type(2))) float v2f;
typedef __attribute__((ext_vector_type(8))) float v8f;

// ---------------------------------------------------------------------------
// Wave32 sum via V_WMMA_F32_16X16X4_F32 (one matrix op collapses all 32 lane
// partials): A 16x4 f32 -> VGPR0 = lanes0-15 @K=0 / lanes16-31 @K=2, VGPR1=0.
// B = ones => D[m][n] = p[lane m] + p[lane m+16]. Lanes 0-15 hold D rows 0-7
// in c[0..7], lanes 16-31 rows 8-15; sum VGPRs then fold +16 lane.
// EXEC all-ones at call sites (full blocks, uniform control flow).
// ---------------------------------------------------------------------------
__device__ __forceinline__ float wave_sum_wmma(float p) {
    v2f a; a.x = p;   a.y = 0.0f;
    v2f b; b.x = 1.0f; b.y = 1.0f;
    v8f c = {};
    c = __builtin_amdgcn_wmma_f32_16x16x4_f32(
            /*neg_a=*/false, a, /*neg_b=*/false, b,
            /*c_mod=*/(short)0, c, /*reuse_a=*/false, /*reuse_b=*/false);
    float s = c[0] + c[1] + c[2] + c[3] + c[4] + c[5] + c[6] + c[7];
    s += __shfl_down(s, 16, 32);
    return s;                      // valid on lane 0 of the wave
}

// Block (256 thr = 8 wave32) reduction; result valid on threadIdx.x == 0.
__device__ __forceinline__ float block_sum(float p, float* lds) {
    float w = wave_sum_wmma(p);
    const int lane = threadIdx.x & 31;
    const int wid  = threadIdx.x >> 5;
    if (lane == 0) lds[wid] = w;
    __syncthreads();
    float s = 0.0f;
    if (threadIdx.x == 0) {
#pragma unroll
        for (int k = 0; k < THREADS / 32; ++k) s += lds[k];
    }
    return s;
}

__device__ __forceinline__ float sq(float v) { return v * v; }

__device__ __forceinline__ float signed_sqrt(float v) {
    float sgn = (v > 0.0f) ? 1.0f : ((v < 0.0f) ? -1.0f : 0.0f);
    return sqrtf(fabsf(v) + 1e-6f) * sgn;
}

// ch = (4*i) % 30 is always even: elem0 can only hit channel 4 (ch==4),
// elem1 only channel 9 (ch==8), elem2 -> 4 (ch==2), elem3 -> 9 (ch==6).
__device__ __forceinline__ float masked_sq_sum(float4 v, unsigned ch) {
    const float m0 = (ch == 4u) ? 1.0f : 0.0f;
    const float m1 = (ch == 8u) ? 1.0f : 0.0f;
    const float m2 = (ch == 2u) ? 1.0f : 0.0f;
    const float m3 = (ch == 6u) ? 1.0f : 0.0f;
    return v.x * v.x * m0 + v.y * v.y * m1 + v.z * v.z * m2 + v.w * v.w * m3;
}

__device__ __forceinline__ unsigned mod30_add(unsigned ch, unsigned d) {
    ch += d;
    return (ch >= 30u) ? ch - 30u : ch;
}

// ---------------------------------------------------------------------------
// Kernel A: LAMDA_NOOBJ * sum(output[...,4]^2 + output[...,9]^2).
// Manually unrolled x4: four independent global_load_b128 issued as a clause
// before any use -> 4 loads in flight per lane (~19 MB device-wide MLP,
// above the ~14 MB latency-BW product needed to saturate 23.3 TB/s HBM).
// Branch-free masks, strength-reduced channel index, exact trip count.
// ---------------------------------------------------------------------------
__global__ void __launch_bounds__(THREADS)
conf_sq_kernel(const float* __restrict__ out, float* __restrict__ part) {
    __shared__ float lds[THREADS / 32];
    const float4* o4 = reinterpret_cast<const float4*>(out);

    const unsigned stride = NBLK_CONF * THREADS;   // float4 units; 4*stride%30==24
    unsigned i  = blockIdx.x * blockDim.x + threadIdx.x;
    unsigned ch = (i * 4u) % 30u;

    float p0 = 0.0f, p1 = 0.0f, p2 = 0.0f, p3 = 0.0f;
#pragma unroll
    for (int k = 0; k < CONF_OUTER; ++k) {
        // group the loads: one vmem clause, 4 b128 in flight
        const float4 v0 = o4[i];
        const float4 v1 = o4[i + stride];
        const float4 v2 = o4[i + 2u * stride];
        const float4 v3 = o4[i + 3u * stride];

        const unsigned chA = ch;
        const unsigned chB = mod30_add(chA, 24u);
        const unsigned chC = mod30_add(chB, 24u);
        const unsigned chD = mod30_add(chC, 24u);

        p0 += masked_sq_sum(v0, chA);
        p1 += masked_sq_sum(v1, chB);
        p2 += masked_sq_sum(v2, chC);
        p3 += masked_sq_sum(v3, chD);

        i += 4u * stride;
        ch = mod30_add(ch, 6u);        // (4*4*stride) % 30 == 6
    }
    const float p = (p0 + p1) + (p2 + p3);

    float s = block_sum(p, lds);
    if (threadIdx.x == 0) part[blockIdx.x] = LAMDA_NOOBJ * s;
}

// ---------------------------------------------------------------------------
// Kernel B: per-target loss, one thread per target (exact fit). All gv[]
// indexing is compile-time constant (selects via cndmask, class gather folded
// into the sum loop) -> no scratch spills. Gathers hit L2 (96 MB < 192 MB)
// after kernel A streamed the tensor.
// ---------------------------------------------------------------------------
__global__ void __launch_bounds__(THREADS)
target_loss_kernel(const float* __restrict__ out, const float* __restrict__ tgt,
                   float* __restrict__ part) {
    __shared__ float lds[THREADS / 32];
    const int t = blockIdx.x * blockDim.x + threadIdx.x;   // < NTGT_N exactly

    const float4* t4 = reinterpret_cast<const float4*>(tgt + t * 8);
    float4 ta = t4[0];   // x, y, w, h
    float4 tb = t4[1];   // gx, gy, cls, bid
    const float x_t = ta.x, y_t = ta.y, w_t = ta.z, h_t = ta.w;
    const int gx = (int)tb.x, gy = (int)tb.y;
    const int cls_t = (int)tb.z, bid = (int)tb.w;

    const float* g = out + (((bid * GRID_N) + gx) * GRID_N + gy) * 30;
    float gv[30];
#pragma unroll
    for (int k = 0; k < 30; ++k) gv[k] = g[k];   // constant indices only

    const float lef_t = x_t - 3.5f * w_t, rig_t = x_t + 3.5f * w_t;
    const float top_t = y_t - 3.5f * h_t, bot_t = y_t + 3.5f * h_t;
    const float area_t = w_t * h_t * 49.0f;

    float iou[2];
#pragma unroll
    for (int j = 0; j < 2; ++j) {
        const float xg = gv[5 * j + 0], yg = gv[5 * j + 1];
        const float wg = gv[5 * j + 2], hg = gv[5 * j + 3];
        const float lef = xg - 3.5f * wg, rig = xg + 3.5f * wg;
        const float top = yg - 3.5f * hg, bot = yg + 3.5f * hg;
        const float w_in = fmaxf(fminf(rig, rig_t) - fmaxf(lef, lef_t), 0.0f);
        const float h_in = fmaxf(fminf(bot, bot_t) - fmaxf(top, top_t), 0.0f);
        const float a_in = w_in * h_in;
        const float a_g  = wg * hg * 49.0f;
        const float a_tot = area_t + a_g - a_in;
        iou[j] = (a_tot > 1e-6f) ? (a_in / a_tot) : 0.0f;
    }
    const bool r1 = iou[1] > iou[0];             // argmax, tie -> box 0

    const float x_r = r1 ? gv[5] : gv[0];
    const float y_r = r1 ? gv[6] : gv[1];
    const float w_r = r1 ? gv[7] : gv[2];
    const float h_r = r1 ? gv[8] : gv[3];
    const float c_r = r1 ? gv[9] : gv[4];

    float loss = LAMDA_COORD * (sq(x_t - x_r) + sq(y_t - y_r));
    loss += LAMDA_SIZE * (sq(signed_sqrt(w_t) - signed_sqrt(w_r)) +
                          sq(signed_sqrt(h_t) - signed_sqrt(h_r)));
    loss += LAMDA_OBJ * sq(c_r - 1.0f) - LAMDA_NOOBJ * sq(c_r);

    float csum = 0.0f, cr = 0.0f;
#pragma unroll
    for (int k = 0; k < 20; ++k) {
        const float c = gv[10 + k];
        csum += c * c;
        cr = (k == cls_t) ? c : cr;              // class gather via cndmask
    }
    loss += csum;
    loss += 1.0f - 2.0f * cr;   // (cls_r-1)^2 - cls_r^2 == 1 - 2*cls_r

    float s = block_sum(loss, lds);
    if (threadIdx.x == 0) part[blockIdx.x] = s;
}

// ---------------------------------------------------------------------------
// Kernel C: deterministic final reduce of all partials -> scalar loss.
// ---------------------------------------------------------------------------
__global__ void __launch_bounds__(THREADS)
final_reduce_kernel(const float* __restrict__ part, int n,
                    float* __restrict__ outp) {
    __shared__ float lds[THREADS / 32];
    float p = 0.0f;
    for (int i = threadIdx.x; i < n; i += THREADS) p += part[i];
    float s = block_sum(p, lds);
    if (threadIdx.x == 0) outp[0] = s;
}

extern "C" void kernel_launch(void* const* d_in, const int* in_sizes, int n_in,
                              void* d_out, int out_size, void* d_ws, size_t ws_size,
                              hipStream_t stream) {
    const float* out = (const float*)d_in[0];   // (16384,7,7,30) f32
    const float* tgt = (const float*)d_in[1];   // (65536,8) f32
    float* part = (float*)d_ws;                 // NBLK_CONF + NBLK_TGT floats

    conf_sq_kernel<<<NBLK_CONF, THREADS, 0, stream>>>(out, part);
    target_loss_kernel<<<NBLK_TGT, THREADS, 0, stream>>>(out, tgt, part + NBLK_CONF);
    final_reduce_kernel<<<1, THREADS, 0, stream>>>(part, NBLK_CONF + NBLK_TGT,
                                                   (float*)d_out);
}